// NeuralNetwork_8014408974469
// MI455X (gfx1250) — compile-verified
//
#include <hip/hip_runtime.h>

#define TPB 256

// ---------------------------------------------------------------------------
// fast transcendentals: v_exp_f32 + v_rcp_f32
// ---------------------------------------------------------------------------
__device__ __forceinline__ float fast_sigmoid(float z) {
    return __builtin_amdgcn_rcpf(1.0f + __expf(-z));
}
__device__ __forceinline__ float fast_tanh(float t) {
    float e = __expf(-2.0f * t);                 // tanh(t) = (1-e^{-2t})/(1+e^{-2t})
    return (1.0f - e) * __builtin_amdgcn_rcpf(1.0f + e);
}

// elementwise plasticity update (matches reference `updated`)
__device__ __forceinline__ float updated_elem(float Wv, float xb, float yb,
                                              float lr, int rule, float exc) {
    float w     = fabsf(Wv);
    float one_w = 1.0f - w;
    float xy    = xb * yb;
    float delta;
    if (rule == 0) {                       // hebb
        delta = lr * xy * one_w;
    } else if (rule == 1) {                // post
        delta = lr * (w * (xb - 1.0f) * yb + one_w * xy);
    } else if (rule == 2) {                // pre
        delta = lr * (w * xb * (yb - 1.0f) + one_w * xy);
    } else {                               // cov
        float d = fast_tanh(-4.0f - 4.0f * fabsf(xb - yb));
        delta = lr * ((d > 0.0f) ? d * one_w : d * w);
    }
    float wn = w + delta;
    return (exc * wn < 0.0f) ? wn * exc : wn;
}

__global__ __launch_bounds__(TPB) void plastic_kernel(
    const float* __restrict__ x,    const float* __restrict__ W1d,
    const float* __restrict__ W1m,  const float* __restrict__ s1d,
    const float* __restrict__ s1m,  const float* __restrict__ excit,
    const float* __restrict__ lrate,const int*   __restrict__ drv,
    const int*   __restrict__ rul,  float* __restrict__ out, int Btot)
{
    // 256 samples/block * 24 floats = 24 KB per matrix tile
    __shared__ __align__(16) float smWd[TPB * 24];
    __shared__ __align__(16) float smWm[TPB * 24];

    const unsigned tid = threadIdx.x;
    const size_t   b   = (size_t)blockIdx.x * TPB + tid;

    // ---- stage W1d / W1m tiles into LDS, coalesced 16B chunks, async ----
    const float* gWd = W1d + (size_t)blockIdx.x * (TPB * 24);
    const float* gWm = W1m + (size_t)blockIdx.x * (TPB * 24);
    const unsigned smWdBase = (unsigned)(unsigned long long)(&smWd[0]);
    const unsigned smWmBase = (unsigned)(unsigned long long)(&smWm[0]);
    const size_t nChunks = (size_t)Btot * 6;                 // 6 x 16B chunks per sample
    const size_t chunk0  = (size_t)blockIdx.x * (TPB * 6) + tid;

#pragma unroll
    for (int it = 0; it < 6; ++it) {
        if (chunk0 + (size_t)(it * TPB) < nChunks) {
            unsigned off = (unsigned)(it * (TPB * 16)) + tid * 16u;
            asm volatile("global_load_async_to_lds_b128 %0, %1, %2"
                         :: "v"(smWdBase + off), "v"(off), "s"(gWd) : "memory");
            asm volatile("global_load_async_to_lds_b128 %0, %1, %2"
                         :: "v"(smWmBase + off), "v"(off), "s"(gWm) : "memory");
        }
    }
    asm volatile("s_wait_asynccnt 0" ::: "memory");
    __syncthreads();

    // output layout (flat, return-tuple order)
    float* o_out  = out;                       // [B,2]
    float* o_Wd   = out + (size_t)2  * Btot;   // [B,3,8]
    float* o_Wm   = out + (size_t)26 * Btot;   // [B,3,8]
    float* o_out1 = out + (size_t)50 * Btot;   // [B,3]
    float* o_out2 = out + (size_t)53 * Btot;   // [B,3]

    if (b < (size_t)Btot) {
        float X1[8], X2[8];
        const float* xp = x + b * 5;
#pragma unroll
        for (int j = 0; j < 5; ++j) { float v = xp[j]; X1[j] = v; X2[j] = v; }
        const float* sdp = s1d + b * 3;
        const float* smp = s1m + b * 3;
#pragma unroll
        for (int j = 0; j < 3; ++j) { X1[5 + j] = sdp[j]; X2[5 + j] = smp[j]; }

        float o1[3], o2[3];

        // ---------------- pass 1: W1d (update when driving_rule==1) --------
        {
            float W[24];
            const float4* p = (const float4*)smWd + tid * 6;
#pragma unroll
            for (int c = 0; c < 6; ++c) {
                float4 q = p[c];
                W[c*4+0] = q.x; W[c*4+1] = q.y; W[c*4+2] = q.z; W[c*4+3] = q.w;
            }
#pragma unroll
            for (int i = 0; i < 3; ++i) {
                float z = 0.0f;
#pragma unroll
                for (int j = 0; j < 8; ++j) z += W[i*8+j] * X1[j];
                o1[i] = fast_sigmoid(z);
            }
#pragma unroll
            for (int i = 0; i < 3; ++i)
#pragma unroll
                for (int j = 0; j < 8; ++j) {
                    int k = i * 8 + j;
                    if (drv[k] == 1)   // wave-uniform branch
                        W[k] = updated_elem(W[k], X1[j], o1[i], lrate[k], rul[k], excit[k]);
                }
            float4* wp = (float4*)smWd + tid * 6;
#pragma unroll
            for (int c = 0; c < 6; ++c) {
                float4 q; q.x=W[c*4+0]; q.y=W[c*4+1]; q.z=W[c*4+2]; q.w=W[c*4+3];
                wp[c] = q;
            }
        }

        // ---------------- pass 2: W1m (update when driving_rule!=1) --------
        {
            float W[24];
            const float4* p = (const float4*)smWm + tid * 6;
#pragma unroll
            for (int c = 0; c < 6; ++c) {
                float4 q = p[c];
                W[c*4+0] = q.x; W[c*4+1] = q.y; W[c*4+2] = q.z; W[c*4+3] = q.w;
            }
#pragma unroll
            for (int i = 0; i < 3; ++i) {
                float z = 0.0f;
#pragma unroll
                for (int j = 0; j < 8; ++j) z += W[i*8+j] * X2[j];
                o2[i] = 0.5f * fast_sigmoid(z) + 0.5f;
            }
#pragma unroll
            for (int i = 0; i < 3; ++i)
#pragma unroll
                for (int j = 0; j < 8; ++j) {
                    int k = i * 8 + j;
                    if (drv[k] != 1)   // wave-uniform branch
                        W[k] = updated_elem(W[k], X2[j], o2[i], lrate[k], rul[k], excit[k]);
                }
            float4* wp = (float4*)smWm + tid * 6;
#pragma unroll
            for (int c = 0; c < 6; ++c) {
                float4 q; q.x=W[c*4+0]; q.y=W[c*4+1]; q.z=W[c*4+2]; q.w=W[c*4+3];
                wp[c] = q;
            }
        }

        // ---- small outputs: direct stores (8/12/12 B per thread) ----------
        o_out[b*2+0] = o1[1] * o1[1];
        o_out[b*2+1] = o1[2] * o1[2];
#pragma unroll
        for (int i = 0; i < 3; ++i) o_out1[b*3+i] = o1[i];
#pragma unroll
        for (int i = 0; i < 3; ++i) o_out2[b*3+i] = o2[i];
    }

    // ---- stream updated weights LDS -> global, coalesced, async -----------
    __syncthreads();
    float* gOd = o_Wd + (size_t)blockIdx.x * (TPB * 24);
    float* gOm = o_Wm + (size_t)blockIdx.x * (TPB * 24);
#pragma unroll
    for (int it = 0; it < 6; ++it) {
        if (chunk0 + (size_t)(it * TPB) < nChunks) {
            unsigned off = (unsigned)(it * (TPB * 16)) + tid * 16u;
            asm volatile("global_store_async_from_lds_b128 %0, %1, %2"
                         :: "v"(off), "v"(smWdBase + off), "s"(gOd) : "memory");
            asm volatile("global_store_async_from_lds_b128 %0, %1, %2"
                         :: "v"(off), "v"(smWmBase + off), "s"(gOm) : "memory");
        }
    }
    asm volatile("s_wait_asynccnt 0" ::: "memory");
}

extern "C" void kernel_launch(void* const* d_in, const int* in_sizes, int n_in,
                              void* d_out, int out_size, void* d_ws, size_t ws_size,
                              hipStream_t stream) {
    const float* x   = (const float*)d_in[0];
    const float* W1d = (const float*)d_in[1];
    const float* W1m = (const float*)d_in[2];
    const float* s1d = (const float*)d_in[3];
    const float* s1m = (const float*)d_in[4];
    const float* exc = (const float*)d_in[5];
    const float* lr  = (const float*)d_in[6];
    const int*   drv = (const int*)d_in[7];
    const int*   rul = (const int*)d_in[8];
    float* out = (float*)d_out;

    const int B = in_sizes[0] / 5;              // x is [B, 5]
    const int blocks = (B + TPB - 1) / TPB;
    plastic_kernel<<<blocks, TPB, 0, stream>>>(x, W1d, W1m, s1d, s1m,
                                               exc, lr, drv, rul, out, B);
}